// SO3onS2_58420145160629
// MI455X (gfx1250) — compile-verified
//
#include <hip/hip_runtime.h>

#define LMAX   8
#define NCOEF  81        // (LMAX+1)^2
#define NTRIP  215
#define CGD    17        // DMAX

typedef float v2f __attribute__((ext_vector_type(2)));
typedef float v8f __attribute__((ext_vector_type(8)));

// d_ws layout (bytes):
//   [0, 8K)      meta: NTRIP * 8 ints {l1, l2, l, K, K4, nt, kofs, cofs}
//   [8K, 72K)    idx table: one u16 per padded-k, value = c1 | (c2<<8)  (column
//                indices into the 81-wide coefficient row; both < 81)
//   [72K, ~2MB)  Cpack: fp32, pre-swizzled into WMMA B-operand lane order
#define META_OFS 0
#define IDX_OFS  8192
#define CP_OFS   73728

// ---------------------------------------------------------------------------
// Setup kernel 1: enumerate the 215 (l1,l2,l) triples in reference order and
// compute per-triple sizes + prefix-sum offsets into the index/Cpack arenas.
// ---------------------------------------------------------------------------
__global__ void so3_meta_kernel(int* __restrict__ meta) {
  if (threadIdx.x != 0 || blockIdx.x != 0) return;
  int t = 0, kofs = 0, cofs = 0;
  for (int l1 = 0; l1 <= LMAX; ++l1) {
    for (int l2 = l1; l2 <= LMAX; ++l2) {
      int ltop = (l1 + l2 < LMAX) ? (l1 + l2) : LMAX;
      for (int l = l2 - l1; l <= ltop; ++l) {
        int d1 = 2 * l1 + 1, d2 = 2 * l2 + 1;
        int K  = d1 * d2;                 // flattened (m1,m2) length
        int K4 = (K + 3) & ~3;            // pad to WMMA K-step granularity
        int nt = (l == LMAX) ? 2 : 1;     // d3 = 17 needs a second N tile
        int* m = meta + t * 8;
        m[0] = l1; m[1] = l2; m[2] = l; m[3] = K;
        m[4] = K4; m[5] = nt; m[6] = kofs; m[7] = cofs;
        kofs += K4;
        cofs += (K4 >> 2) * nt * 64;      // 64 floats per (step, ntile)
        ++t;
      }
    }
  }
}

// ---------------------------------------------------------------------------
// Setup kernel 2: build (a) the u16 k -> (c1,c2) column index table and
// (b) Cpack: the CG tensor zero-padded and swizzled into the exact per-lane
// layout of the V_WMMA_F32_16X16X4_F32 B operand (4x16, K split lo/hi half-
// wave, 2 consecutive K per lane -> one b64 load in the hot loop).
// ---------------------------------------------------------------------------
__global__ void so3_pack_kernel(const float* __restrict__ cg,
                                const int* __restrict__ meta,
                                unsigned short* __restrict__ itab,
                                float* __restrict__ cpack) {
  const int t = blockIdx.x;
  const int* m = meta + t * 8;
  const int l1 = m[0], l2 = m[1], l = m[2];
  const int K = m[3], K4 = m[4], nt = m[5], kofs = m[6], cofs = m[7];
  const int d2 = 2 * l2 + 1, d3 = 2 * l + 1;
  const int b1 = l1 * l1, b2 = l2 * l2;

  for (int k = threadIdx.x; k < K4; k += blockDim.x) {
    unsigned c1 = 0, c2 = 0;
    if (k < K) { int m1 = k / d2; int m2 = k - m1 * d2; c1 = b1 + m1; c2 = b2 + m2; }
    itab[kofs + k] = (unsigned short)(c1 | (c2 << 8));
  }

  const int total = (K4 >> 2) * nt * 64;
  const float* cgt = cg + (size_t)t * (CGD * CGD * CGD);
  for (int i = threadIdx.x; i < total; i += blockDim.x) {
    int e  = i & 1;             // which of the 2 K's this lane holds
    int j  = (i >> 1) & 31;     // lane
    int sn = i >> 6;            // (step, ntile) pair
    int n  = sn % nt, s = sn / nt;
    int h  = j >> 4;            // half-wave -> K base 0 or 2
    int k  = 4 * s + 2 * h + e;
    int col = n * 16 + (j & 15);
    float val = 0.f;
    if (k < K && col < d3) {
      int m1 = k / d2, m2 = k - m1 * d2;
      val = cgt[(m1 * CGD + m2) * CGD + col];
    }
    cpack[cofs + i] = val;
  }
}

// ---------------------------------------------------------------------------
// Main kernel: 4 waves / block, each wave owns a 32-row M tile (2 WMMA
// accumulators sharing every B operand load). All waves stream the same
// triple list -> Cpack stays hot in WGP$/L2.
// ---------------------------------------------------------------------------
__launch_bounds__(128)
__global__ void so3_main_kernel(const float* __restrict__ x,
                                const int* __restrict__ meta,
                                const unsigned short* __restrict__ itab,
                                const float* __restrict__ cpack,
                                float* __restrict__ out) {
  __shared__ float xs[128 * NCOEF];            // 41.5 KB x tile

  const int tid = threadIdx.x;
  const size_t brow0 = (size_t)blockIdx.x * 128;

  // rows are contiguous: flat coalesced copy of 128*81 floats
  const float* xsrc = x + brow0 * NCOEF;
  for (int i = tid; i < 128 * NCOEF; i += 128) xs[i] = xsrc[i];
  __syncthreads();

  const int wv = tid >> 5, lane = tid & 31;
  const int h = lane >> 4;                     // half-wave (K 0/1 vs 2/3)
  const int nn = lane & 15;                    // N index / M sub-row
  const int wrow = wv * 32;                    // wave's row base in block
  const float* xr0 = xs + (wrow + nn) * NCOEF;        // A rows, subtile 0
  const float* xr1 = xs + (wrow + 16 + nn) * NCOEF;   // A rows, subtile 1

  for (int t = 0; t < NTRIP; ++t) {
    const int* m = meta + t * 8;
    const int l = m[2], nt = m[5], kofs = m[6], cofs = m[7];
    const int steps = m[4] >> 2;
    const unsigned short* it = itab + kofs;
    const float* cp = cpack + cofs;
    const int lb = l * l;

    v8f a0 = {0.f,0.f,0.f,0.f,0.f,0.f,0.f,0.f};   // G[rows 0..15,  m 0..15]
    v8f a1 = {0.f,0.f,0.f,0.f,0.f,0.f,0.f,0.f};   // G[rows 16..31, m 0..15]
    v8f b0 = {0.f,0.f,0.f,0.f,0.f,0.f,0.f,0.f};   // l==8: m 16..31 tile
    v8f b1 = {0.f,0.f,0.f,0.f,0.f,0.f,0.f,0.f};

    if (nt == 1) {
      for (int s = 0; s < steps; ++s) {
        unsigned v = *(const unsigned*)(it + 4 * s + 2 * h);
        int c1a = v & 255, c2a = (v >> 8) & 255;
        int c1b = (v >> 16) & 255, c2b = v >> 24;
        v2f A0 = { xr0[c1a] * xr0[c2a], xr0[c1b] * xr0[c2b] };
        v2f A1 = { xr1[c1a] * xr1[c2a], xr1[c1b] * xr1[c2b] };
        v2f Bv = *(const v2f*)(cp + ((size_t)s * 32 + lane) * 2);
        a0 = __builtin_amdgcn_wmma_f32_16x16x4_f32(false, A0, false, Bv,
                                                   (short)0, a0, false, false);
        a1 = __builtin_amdgcn_wmma_f32_16x16x4_f32(false, A1, false, Bv,
                                                   (short)0, a1, false, false);
      }
    } else {
      for (int s = 0; s < steps; ++s) {
        unsigned v = *(const unsigned*)(it + 4 * s + 2 * h);
        int c1a = v & 255, c2a = (v >> 8) & 255;
        int c1b = (v >> 16) & 255, c2b = v >> 24;
        v2f A0 = { xr0[c1a] * xr0[c2a], xr0[c1b] * xr0[c2b] };
        v2f A1 = { xr1[c1a] * xr1[c2a], xr1[c1b] * xr1[c2b] };
        v2f B0v = *(const v2f*)(cp + ((size_t)(s * 2 + 0) * 32 + lane) * 2);
        v2f B1v = *(const v2f*)(cp + ((size_t)(s * 2 + 1) * 32 + lane) * 2);
        a0 = __builtin_amdgcn_wmma_f32_16x16x4_f32(false, A0, false, B0v,
                                                   (short)0, a0, false, false);
        a1 = __builtin_amdgcn_wmma_f32_16x16x4_f32(false, A1, false, B0v,
                                                   (short)0, a1, false, false);
        b0 = __builtin_amdgcn_wmma_f32_16x16x4_f32(false, A0, false, B1v,
                                                   (short)0, b0, false, false);
        b1 = __builtin_amdgcn_wmma_f32_16x16x4_f32(false, A1, false, B1v,
                                                   (short)0, b1, false, false);
      }
    }

    // Epilogue: beta[b] = sum_m G[b,m] * F3[b,m].
    // D layout: lane holds (M = r + 8*h, N = nn) per VGPR r; reduce over N
    // with an xor-shuffle tree inside each 16-lane half.
#pragma unroll
    for (int r = 0; r < 8; ++r) {
      const int row = r + 8 * h;
      float v0 = a0[r] * xs[(wrow + row) * NCOEF + lb + nn];
      float v1 = a1[r] * xs[(wrow + 16 + row) * NCOEF + lb + nn];
      if (nt == 2 && nn == 0) {       // the single valid column of tile 1 (m=16)
        v0 += b0[r] * xs[(wrow + row) * NCOEF + lb + 16];
        v1 += b1[r] * xs[(wrow + 16 + row) * NCOEF + lb + 16];
      }
      v0 += __shfl_xor(v0, 1, 32);  v1 += __shfl_xor(v1, 1, 32);
      v0 += __shfl_xor(v0, 2, 32);  v1 += __shfl_xor(v1, 2, 32);
      v0 += __shfl_xor(v0, 4, 32);  v1 += __shfl_xor(v1, 4, 32);
      v0 += __shfl_xor(v0, 8, 32);  v1 += __shfl_xor(v1, 8, 32);
      if (nn == 0) {                  // lanes 0 and 16 hold the two half sums
        out[(brow0 + wrow + row) * NTRIP + t] = v0;
        out[(brow0 + wrow + 16 + row) * NTRIP + t] = v1;
      }
    }
  }
}

// ---------------------------------------------------------------------------
extern "C" void kernel_launch(void* const* d_in, const int* in_sizes, int n_in,
                              void* d_out, int out_size, void* d_ws, size_t ws_size,
                              hipStream_t stream) {
  (void)n_in; (void)out_size; (void)ws_size;
  const float* x  = (const float*)d_in[0];
  const float* cg = (const float*)d_in[1];
  float* out = (float*)d_out;
  const int B = in_sizes[0] / NCOEF;           // 131072

  int*            meta  = (int*)((char*)d_ws + META_OFS);
  unsigned short* itab  = (unsigned short*)((char*)d_ws + IDX_OFS);
  float*          cpack = (float*)((char*)d_ws + CP_OFS);

  so3_meta_kernel<<<1, 32, 0, stream>>>(meta);
  so3_pack_kernel<<<NTRIP, 256, 0, stream>>>(cg, meta, itab, cpack);
  so3_main_kernel<<<B / 128, 128, 0, stream>>>(x, meta, itab, cpack, out);
}